// BlockFFT_5849745457162
// MI455X (gfx1250) — compile-verified
//
#include <hip/hip_runtime.h>
#include <hip/hip_bf16.h>

typedef __attribute__((ext_vector_type(2))) float v2f;
typedef __attribute__((ext_vector_type(4))) float v4f;
typedef __attribute__((ext_vector_type(8))) float v8f;

#define NFFT    4096
#define THREADS 256
#define PI_F    3.14159265358979323846f

// Radix-16, 3-stage FFT of length 4096 per (b,h) vector.
// D = M(16x16) * X(16x16) complex, done as 4 real 16x16x16 matmuls,
// each real matmul = 4 chained V_WMMA_F32_16X16X4_F32.
__global__ __launch_bounds__(THREADS) void fft4096_r16_kernel(
    const float* __restrict__ xr, const float* __restrict__ xi,
    const float* __restrict__ mat16, float* __restrict__ out, int H)
{
  __shared__ float sr[NFFT];
  __shared__ float si[NFFT];
  __shared__ float mr[256];
  __shared__ float mi[256];

  const int bh   = blockIdx.x;
  const int h    = bh % H;                 // x is (B,H,N): flat = (b*H+h)*N
  const int tid  = threadIdx.x;
  const int lane = tid & 31;
  const int wave = tid >> 5;               // 8 waves
  const int hi   = lane >> 4;              // lane half (K/M split in WMMA layouts)
  const int lo   = lane & 15;

  const size_t vbase = (size_t)bh * NFFT;

  // ---- load the 4096-complex vector (vectorized B128) ----
  {
    const v4f* xr4 = (const v4f*)(xr + vbase);
    const v4f* xi4 = (const v4f*)(xi + vbase);
    v4f* sr4 = (v4f*)sr;
    v4f* si4 = (v4f*)si;
#pragma unroll
    for (int it = 0; it < NFFT / 4 / THREADS; ++it) {
      int k = it * THREADS + tid;
      sr4[k] = xr4[k];
      si4[k] = xi4[k];
    }
  }
  // ---- load the h-specific 16x16 DFT matrix; (H,16,16,2) -> one v2f/thread
  {
    v2f m = ((const v2f*)mat16)[(size_t)h * 256 + tid];
    mr[tid] = m[0];
    mi[tid] = m[1];
  }
  __syncthreads();

  // ---- A fragments (held in registers for all 3 stages) ----
  // A(16x4) layout: lane L holds M[row=L%16][K = kk*4 + 2*(L>>4) + v], v=0,1
  v2f aR[4], aI[4], aNI[4];
#pragma unroll
  for (int kk = 0; kk < 4; ++kk) {
#pragma unroll
    for (int v = 0; v < 2; ++v) {
      int ko = kk * 4 + 2 * hi + v;
      float re = mr[lo * 16 + ko];
      float im = mi[lo * 16 + ko];
      aR[kk][v]  = re;
      aI[kk][v]  = im;
      aNI[kk][v] = -im;   // f32 WMMA has no A/B NEG modifier -> pre-negate Mi
    }
  }

  // Complex 16x16 matmul: (Dr + iDi) = (Mr + iMi)(Br + iBi)
  auto cmatmul = [&](const v2f bR[4], const v2f bI[4], v8f& dR, v8f& dI) {
    dR = {}; dI = {};
#pragma unroll
    for (int kk = 0; kk < 4; ++kk)
      dR = __builtin_amdgcn_wmma_f32_16x16x4_f32(false, aR[kk],  false, bR[kk], (short)0, dR, false, false);
#pragma unroll
    for (int kk = 0; kk < 4; ++kk)
      dR = __builtin_amdgcn_wmma_f32_16x16x4_f32(false, aNI[kk], false, bI[kk], (short)0, dR, false, false);
#pragma unroll
    for (int kk = 0; kk < 4; ++kk)
      dI = __builtin_amdgcn_wmma_f32_16x16x4_f32(false, aI[kk],  false, bR[kk], (short)0, dI, false, false);
#pragma unroll
    for (int kk = 0; kk < 4; ++kk)
      dI = __builtin_amdgcn_wmma_f32_16x16x4_f32(false, aR[kk],  false, bI[kk], (short)0, dI, false, false);
  };

  // ================= Stage A: rows stride 256, tile = 16 columns ===========
  // In-place per tile (read & write the same 16x16 column footprint).
  for (int t = wave; t < 16; t += 8) {
    const int col = t * 16 + lo;
    v2f bR[4], bI[4];
#pragma unroll
    for (int kk = 0; kk < 4; ++kk) {
#pragma unroll
      for (int v = 0; v < 2; ++v) {
        int k = kk * 4 + 2 * hi + v;           // B(4x16): K split like A
        bR[kk][v] = sr[k * 256 + col];
        bI[kk][v] = si[k * 256 + col];
      }
    }
    v8f dR, dI;
    cmatmul(bR, bI, dR, dI);

    // twiddle W4096^(col*r0), r0 = v + 8*hi; rotation recurrence over v.
    // All angles are within [-pi, pi] -> branch-free hardware sin/cos.
    const float tb = (-2.0f * PI_F / 4096.0f) * (float)col;
    float s0, c0, s1, c1;
    __sincosf(tb * (float)(8 * hi), &s0, &c0);
    __sincosf(tb, &s1, &c1);
    float wr = c0, wi = s0;
#pragma unroll
    for (int v = 0; v < 8; ++v) {
      float yr = dR[v] * wr - dI[v] * wi;
      float yi = dR[v] * wi + dI[v] * wr;
      const int r0 = v + 8 * hi;
      sr[r0 * 256 + col] = yr;
      si[r0 * 256 + col] = yi;
      float wr2 = wr * c1 - wi * s1;
      wi = wr * s1 + wi * c1;
      wr = wr2;
    }
  }
  __syncthreads();   // stage B reads whole rows written by all tiles of stage A

  // ========== Stages B + C fused: row t owned by one wave throughout =======
  for (int t = wave; t < 16; t += 8) {
    const int rowb = t * 256;   // row r0 = t

    // ---- Stage B: Y2[r1,n1] = sum_o1 M[r1,o1] * V1[t, o1*16 + n1] ----
    v2f bR[4], bI[4];
#pragma unroll
    for (int kk = 0; kk < 4; ++kk) {
#pragma unroll
      for (int v = 0; v < 2; ++v) {
        int k = kk * 4 + 2 * hi + v;
        bR[kk][v] = sr[rowb + k * 16 + lo];
        bI[kk][v] = si[rowb + k * 16 + lo];
      }
    }
    v8f dR, dI;
    cmatmul(bR, bI, dR, dI);

    // twiddle W256^(n1*r1), n1 = lo, r1 = v + 8*hi; store transposed in-row
    const float tb = (-2.0f * PI_F / 256.0f) * (float)lo;
    float s0, c0, s1, c1;
    __sincosf(tb * (float)(8 * hi), &s0, &c0);
    __sincosf(tb, &s1, &c1);
    float wr = c0, wi = s0;
#pragma unroll
    for (int v = 0; v < 8; ++v) {
      float yr = dR[v] * wr - dI[v] * wi;
      float yi = dR[v] * wi + dI[v] * wr;
      const int r1 = v + 8 * hi;
      sr[rowb + lo * 16 + r1] = yr;    // element (row=n1, col=r1)
      si[rowb + lo * 16 + r1] = yi;
      float wr2 = wr * c1 - wi * s1;
      wi = wr * s1 + wi * c1;
      wr = wr2;
    }
    // same wave wrote the whole row; per-wave LDS ordering -> no barrier

    // ---- Stage C: Y3[r2,r1] = sum_o2 M[r2,o2] * X[o2*16 + r1] ----
#pragma unroll
    for (int kk = 0; kk < 4; ++kk) {
#pragma unroll
      for (int v = 0; v < 2; ++v) {
        int k = kk * 4 + 2 * hi + v;
        bR[kk][v] = sr[rowb + k * 16 + lo];
        bI[kk][v] = si[rowb + k * 16 + lo];
      }
    }
    cmatmul(bR, bI, dR, dI);

    // digit-reversed output j = r2*256 + r1*16 + r0; out is (B,H,N,2) -> float2
    v2f* out2 = (v2f*)out;
#pragma unroll
    for (int v = 0; v < 8; ++v) {
      const int r2 = v + 8 * hi;
      const int j  = r2 * 256 + lo * 16 + t;
      v2f val;
      val[0] = dR[v];
      val[1] = dI[v];
      out2[vbase + j] = val;
    }
  }
}

extern "C" void kernel_launch(void* const* d_in, const int* in_sizes, int n_in,
                              void* d_out, int out_size, void* d_ws, size_t ws_size,
                              hipStream_t stream) {
  (void)n_in; (void)out_size; (void)d_ws; (void)ws_size;
  const float* xr  = (const float*)d_in[0];   // x_real (B,H,N)
  const float* xi  = (const float*)d_in[1];   // x_imag (B,H,N)
  const float* m16 = (const float*)d_in[5];   // mat_16 (H,16,16,2); mat_2/4/8 unused for N=4096
  float* out = (float*)d_out;

  const int H    = in_sizes[5] / 512;         // H*16*16*2
  const int nvec = in_sizes[0] / NFFT;        // B*H workgroups

  fft4096_r16_kernel<<<nvec, THREADS, 0, stream>>>(xr, xi, m16, out, H);
}